// MoERouter_34703335752415
// MI455X (gfx1250) — compile-verified
//
#include <hip/hip_runtime.h>
#include <hip/hip_bf16.h>

#define HIDDEN 4096
#define NEXP 8
#define TPB   64      // tokens per block
#define KTILE 64
#define NT    (HIDDEN / KTILE)   // 64 K-tiles
#define A_STRIDE 66   // padded row stride for A tile (bank-conflict free fragment reads)
#define S_STRIDE 19   // padded row stride for score scratch

typedef float v2f __attribute__((ext_vector_type(2)));
typedef float v8f __attribute__((ext_vector_type(8)));

__global__ __launch_bounds__(128) void moe_router_wmma_kernel(
    const float* __restrict__ x,      // [tokens, HIDDEN]
    const float* __restrict__ W,      // [NEXP, HIDDEN]
    float*       __restrict__ out_w,  // [tokens, 2]
    int*         __restrict__ out_i,  // [tokens, 2]
    int tokens)
{
  __shared__ float As[2][TPB][A_STRIDE];   // 2 * 64 * 66 * 4B = 33.8 KB
  __shared__ float Bs[2][KTILE][16];       // 2 * 64 * 16 * 4B =  8.0 KB (experts padded to 16)
  __shared__ float Sc[TPB][S_STRIDE];      // 64 * 19 * 4B     =  4.9 KB

  const int t         = threadIdx.x;       // 0..127
  const int tokenBase = blockIdx.x * TPB;
  const int wave      = t >> 5;            // 0..3
  const int lane      = t & 31;
  const int half      = lane >> 4;         // 0: K-lo pair, 1: K-hi pair
  const int l         = lane & 15;         // row (A) / column (B,C,D) index
  const int tb        = wave * 16;         // this wave's token sub-block

  // Zero the padded expert columns (8..15) of both B buffers once.
  for (int idx = t; idx < 2 * KTILE * 8; idx += 128) {
    int b = idx >> 9;              // /512
    int r = (idx >> 3) & 63;
    Bs[b][r][8 + (idx & 7)] = 0.0f;
  }

  float4 ra[8];   // staged A tile: 64x64 f32 / 128 threads = 8 float4 each
  float  rb[4];   // staged B tile: 64x8  f32 / 128 threads = 4 floats each

  auto loadRegs = [&](int kt) {
    const int k0 = kt * KTILE;
#pragma unroll
    for (int i = 0; i < 8; ++i) {
      int lin = t + i * 128;             // float4 index within tile
      int row = lin >> 4;                // 16 float4 per 64-float row
      int c4  = lin & 15;
      ra[i] = *(const float4*)(x + (size_t)(tokenBase + row) * HIDDEN + k0 + c4 * 4);
    }
#pragma unroll
    for (int i = 0; i < 4; ++i) {
      int idx = t + i * 128;             // 0..511
      int e   = idx >> 6;                // expert
      int k   = idx & 63;
      rb[i] = W[(size_t)e * HIDDEN + k0 + k];
    }
  };

  auto storeLDS = [&](int b) {
#pragma unroll
    for (int i = 0; i < 8; ++i) {
      int lin = t + i * 128;
      int row = lin >> 4;
      int c4  = lin & 15;
      *(float2*)&As[b][row][c4 * 4]     = make_float2(ra[i].x, ra[i].y);
      *(float2*)&As[b][row][c4 * 4 + 2] = make_float2(ra[i].z, ra[i].w);
    }
#pragma unroll
    for (int i = 0; i < 4; ++i) {
      int idx = t + i * 128;
      int e   = idx >> 6;
      int k   = idx & 63;
      Bs[b][k][e] = rb[i];
    }
  };

  v8f acc = {};   // 16x16 f32 scores accumulator (wave-level)

  // Prologue: stage tile 0.
  loadRegs(0);
  storeLDS(0);

  for (int i = 0; i < NT; ++i) {
    const int cur = i & 1;
    const int nxt = cur ^ 1;

    if (i + 1 < NT) loadRegs(i + 1);   // issue next tile's global loads early

    __syncthreads();                   // As/Bs[cur] ready; prior reads of [nxt] done

    // 16 x V_WMMA_F32_16X16X4_F32 over this 64-deep K tile.
#pragma unroll
    for (int kk = 0; kk < KTILE; kk += 4) {
      // A fragment (16x4): lane row = l, K pair = kk + 2*half
      v2f a = *(const v2f*)&As[cur][tb + l][kk + 2 * half];
      // B fragment (4x16): column = l, rows kk+2*half, kk+2*half+1
      v2f b;
      b.x = Bs[cur][kk + 2 * half][l];
      b.y = Bs[cur][kk + 2 * half + 1][l];
      acc = __builtin_amdgcn_wmma_f32_16x16x4_f32(
          /*neg_a=*/false, a, /*neg_b=*/false, b,
          /*c_mod=*/(short)0, acc, /*reuse_a=*/false, /*reuse_b=*/false);
    }

    if (i + 1 < NT) storeLDS(nxt);     // safe: [nxt] last read before this iter's sync
  }

  // Scatter wave accumulators to LDS: row = token-in-block, col = expert.
#pragma unroll
  for (int r = 0; r < 8; ++r) {
    Sc[tb + half * 8 + r][l] = acc[r];
  }
  __syncthreads();

  // One thread per token: softmax + top-2 + renormalize.
  if (t < TPB) {
    const int token = tokenBase + t;
    float s[NEXP];
#pragma unroll
    for (int e = 0; e < NEXP; ++e) s[e] = Sc[t][e];

    int   i0 = 0;
    float m0 = s[0];
#pragma unroll
    for (int e = 1; e < NEXP; ++e) {
      if (s[e] > m0) { m0 = s[e]; i0 = e; }   // strict >: lowest index wins ties (JAX top_k)
    }
    int   i1 = -1;
    float m1 = -3.402823466e+38f;
#pragma unroll
    for (int e = 0; e < NEXP; ++e) {
      if (e != i0 && s[e] > m1) { m1 = s[e]; i1 = e; }
    }

    // Renormalized top-2 softmax: softmax sum cancels.
    float e1 = expf(m1 - m0);
    float w0 = 1.0f / (1.0f + e1);
    float w1 = 1.0f - w0;

    out_w[token * 2 + 0] = w0;
    out_w[token * 2 + 1] = w1;
    out_i[token * 2 + 0] = i0;
    out_i[token * 2 + 1] = i1;
  }
}

extern "C" void kernel_launch(void* const* d_in, const int* in_sizes, int n_in,
                              void* d_out, int out_size, void* d_ws, size_t ws_size,
                              hipStream_t stream) {
  const float* x = (const float*)d_in[0];
  const float* W = (const float*)d_in[1];
  const int tokens = in_sizes[0] / HIDDEN;          // 4*4096 = 16384

  float* out_w = (float*)d_out;                      // [tokens, 2] weights
  int*   out_i = (int*)d_out + (size_t)2 * tokens;   // [tokens, 2] indices (int32)

  const int blocks = tokens / TPB;                   // 256
  hipLaunchKernelGGL(moe_router_wmma_kernel, dim3(blocks), dim3(128), 0, stream,
                     x, W, out_w, out_i, tokens);
}